// TransformerMapper_32006096290581
// MI455X (gfx1250) — compile-verified
//
#include <hip/hip_runtime.h>
#include <hip/hip_bf16.h>
#include <math.h>

// ---------------- types ----------------
typedef __attribute__((ext_vector_type(2))) float v2f;
typedef __attribute__((ext_vector_type(8))) float v8f;

#define DD     768
#define HH     12
#define DH     64
#define NN     20
#define BB     32
#define MM     16384
#define KK     32
#define DFF    3072
#define CLIPL  10
#define QKVW   2304

#define MODE_STORE 0
#define MODE_BIAS  1
#define MODE_RESID 2
#define MODE_GELU  3

// Async copy of one 16-byte segment from global memory into LDS (gfx1250).
__device__ __forceinline__ void async_b128_to_lds(const float* gp, void* lp) {
    unsigned loff = (unsigned)(uintptr_t)lp;     // LDS byte offset (addr[31:0])
    asm volatile("global_load_async_to_lds_b128 %0, %1, off"
                 :
                 : "v"(loff), "v"(gp)
                 : "memory");
}
__device__ __forceinline__ void wait_async0() {
    asm volatile("s_wait_asynccnt 0x0" ::: "memory");
}

// ---------------- generic fp32 WMMA GEMM: C[M,N] = A[M,K] @ B[K,N] (+epilogue) ----------------
// block = 128 threads (4 waves); each wave owns one 16x16 N-subtile of a 16x64 macro tile.
// A panel (16 x 64 per K-chunk) is staged to LDS via async-to-LDS loads and shared by all 4 waves.
__global__ void k_gemm_f32_wmma(const float* __restrict__ A, const float* __restrict__ B,
                                const float* __restrict__ bias, const float* __restrict__ resid,
                                float* __restrict__ C, int Mr, int Kd, int Nd, int mode) {
    int tid = threadIdx.x;          // 0..127
    int wid = tid >> 5;             // wave 0..3
    int lane = tid & 31;
    int lr = lane & 15;
    int hi = lane >> 4;
    int m0 = blockIdx.y * 16;
    int n0 = blockIdx.x * 64 + wid * 16;

    __shared__ __align__(16) float a_sh[16][64];

    const float* bcol = B + (size_t)(2 * hi) * Nd + n0 + lr;   // B(2*hi, n0+lr)
    v8f acc = {};

    for (int kc = 0; kc < Kd; kc += 64) {
        // stage A[m0:m0+16, kc:kc+64] into LDS: 256 x 16B segments
        for (int seg = tid; seg < 256; seg += 128) {
            int r = seg >> 4;
            int e4 = (seg & 15) * 4;
            async_b128_to_lds(A + (size_t)(m0 + r) * Kd + kc + e4, &a_sh[r][e4]);
        }
        __builtin_prefetch(bcol + (size_t)(kc + 64) * Nd, 0, 1);
        wait_async0();
        __syncthreads();

#pragma unroll 4
        for (int k0 = 0; k0 < 64; k0 += 4) {
            v2f a = *(const v2f*)(&a_sh[lr][k0 + 2 * hi]);
            v2f bf;
            bf.x = bcol[(size_t)(kc + k0) * Nd];
            bf.y = bcol[(size_t)(kc + k0 + 1) * Nd];
            acc = __builtin_amdgcn_wmma_f32_16x16x4_f32(false, a, false, bf, (short)0, acc,
                                                        false, false);
        }
        __syncthreads();
    }

    int col = n0 + lr;
#pragma unroll
    for (int v = 0; v < 8; ++v) {
        int rowm = m0 + v + 8 * hi;
        size_t idx = (size_t)rowm * Nd + col;
        float val = acc[v];
        if (mode == MODE_BIAS) {
            val += bias[col];
        } else if (mode == MODE_RESID) {
            val += resid[idx];
        } else if (mode == MODE_GELU) {
            float x = val;
            val = 0.5f * x * (1.0f + tanhf(0.7978845608028654f * (x + 0.044715f * x * x * x)));
        }
        C[idx] = val;
    }
}

// ---------------- build seq: rows 0..9 from h, rows 10..19 from prefix ----------------
__global__ void k_build_seq(const float* __restrict__ h, const float* __restrict__ prefix,
                            float* __restrict__ seq) {
    int idx = blockIdx.x * 256 + threadIdx.x;
    if (idx >= BB * NN * DD) return;
    int b = idx / (NN * DD);
    int rem = idx % (NN * DD);
    int t = rem / DD;
    int d = rem % DD;
    seq[idx] = (t < CLIPL) ? h[(size_t)b * (CLIPL * DD) + t * DD + d]
                           : prefix[(size_t)(t - CLIPL) * DD + d];
}

// ---------------- layernorm: one 256-thread block per row of 768 ----------------
__global__ void k_ln(const float* __restrict__ x, const float* __restrict__ g,
                     const float* __restrict__ b, float* __restrict__ y) {
    int row = blockIdx.x;
    int tid = threadIdx.x;
    const float* xr = x + (size_t)row * DD;
    __shared__ float red[256];
    float s = 0.f, s2 = 0.f;
    for (int d = tid; d < DD; d += 256) { float v = xr[d]; s += v; s2 += v * v; }
    red[tid] = s; __syncthreads();
    for (int off = 128; off > 0; off >>= 1) { if (tid < off) red[tid] += red[tid + off]; __syncthreads(); }
    float mean = red[0] / (float)DD; __syncthreads();
    red[tid] = s2; __syncthreads();
    for (int off = 128; off > 0; off >>= 1) { if (tid < off) red[tid] += red[tid + off]; __syncthreads(); }
    float var = red[0] / (float)DD - mean * mean;
    float rs = rsqrtf(var + 1e-5f);
    float* yr = y + (size_t)row * DD;
    for (int d = tid; d < DD; d += 256) yr[d] = (xr[d] - mean) * rs * g[d] + b[d];
}

// ---------------- final layernorm + slice seq[:,10:,:] -> out ----------------
__global__ void k_final_ln(const float* __restrict__ seq, const float* __restrict__ g,
                           const float* __restrict__ bb, float* __restrict__ out) {
    int row = blockIdx.x;           // 0..BB*10-1
    int b = row / CLIPL;
    int t = row % CLIPL;
    int tid = threadIdx.x;
    const float* xr = seq + ((size_t)b * NN + CLIPL + t) * DD;
    __shared__ float red[256];
    float s = 0.f, s2 = 0.f;
    for (int d = tid; d < DD; d += 256) { float v = xr[d]; s += v; s2 += v * v; }
    red[tid] = s; __syncthreads();
    for (int off = 128; off > 0; off >>= 1) { if (tid < off) red[tid] += red[tid + off]; __syncthreads(); }
    float mean = red[0] / (float)DD; __syncthreads();
    red[tid] = s2; __syncthreads();
    for (int off = 128; off > 0; off >>= 1) { if (tid < off) red[tid] += red[tid + off]; __syncthreads(); }
    float var = red[0] / (float)DD - mean * mean;
    float rs = rsqrtf(var + 1e-5f);
    float* yr = out + (size_t)row * DD;
    for (int d = tid; d < DD; d += 256) yr[d] = (xr[d] - mean) * rs * g[d] + bb[d];
}

// ---------------- local causal attention (no memory), one block per (b,h) ----------------
__global__ void k_attn(const float* __restrict__ qkv, float* __restrict__ ao) {
    int bh = blockIdx.x;
    int b = bh / HH;
    int h = bh % HH;
    int tid = threadIdx.x;   // 128 threads
    __shared__ float q[NN][DH], kk[NN][DH], vv[NN][DH], sim[NN][NN];
    for (int t = tid; t < NN * DH; t += 128) {
        int i = t >> 6, d = t & 63;
        size_t base = ((size_t)b * NN + i) * QKVW + h * DH + d;
        q[i][d] = qkv[base];
        kk[i][d] = qkv[base + DD];
        vv[i][d] = qkv[base + 2 * DD];
    }
    __syncthreads();
    for (int t = tid; t < NN * NN; t += 128) {
        int i = t / NN, j = t % NN;
        float s;
        if (j > i) s = -1e9f;
        else { s = 0.f; for (int d = 0; d < DH; ++d) s += q[i][d] * kk[j][d]; s *= 0.125f; }
        sim[i][j] = s;
    }
    __syncthreads();
    if (tid < NN) {
        int i = tid;
        float mx = -3e38f;
        for (int j = 0; j < NN; ++j) mx = fmaxf(mx, sim[i][j]);
        float sum = 0.f;
        for (int j = 0; j < NN; ++j) { float e = expf(sim[i][j] - mx); sim[i][j] = e; sum += e; }
        float inv = 1.f / sum;
        for (int j = 0; j < NN; ++j) sim[i][j] *= inv;
    }
    __syncthreads();
    for (int t = tid; t < NN * DH; t += 128) {
        int i = t >> 6, d = t & 63;
        float s = 0.f;
        for (int j = 0; j <= i; ++j) s += sim[i][j] * vv[j][d];
        ao[((size_t)b * NN + i) * DD + h * DH + d] = s;
    }
}

// ---------------- memory-augmented attention, one block per (b,h) ----------------
__global__ void k_attn_mem(const float* __restrict__ qkv, const float* __restrict__ ts,
                           const int* __restrict__ ti, const float* __restrict__ memv,
                           float* __restrict__ ao) {
    int bh = blockIdx.x;
    int b = bh / HH;
    int h = bh % HH;
    int tid = threadIdx.x;   // 128 threads
    __shared__ float q[NN][DH], kk[NN][DH], vv[NN][DH], sim[NN][NN];
    __shared__ float am[NN][KK];
    __shared__ int ami[NN][KK];
    for (int t = tid; t < NN * DH; t += 128) {
        int i = t >> 6, d = t & 63;
        size_t base = ((size_t)b * NN + i) * QKVW + h * DH + d;
        q[i][d] = qkv[base];
        kk[i][d] = qkv[base + DD];
        vv[i][d] = qkv[base + 2 * DD];
    }
    for (int t = tid; t < NN * KK; t += 128) {
        int i = t >> 5, k = t & 31;
        size_t base = ((size_t)b * (HH * NN) + h * NN + i) * KK + k;
        am[i][k] = ts[base];
        ami[i][k] = ti[base];
    }
    __syncthreads();
    for (int t = tid; t < NN * NN; t += 128) {
        int i = t / NN, j = t % NN;
        float s;
        if (j > i) s = -1e9f;
        else { s = 0.f; for (int d = 0; d < DH; ++d) s += q[i][d] * kk[j][d]; s *= 0.125f; }
        sim[i][j] = s;
    }
    __syncthreads();
    if (tid < NN) {
        int i = tid;
        float mx = -3e38f;
        for (int j = 0; j < NN; ++j) mx = fmaxf(mx, sim[i][j]);
        for (int k = 0; k < KK; ++k) mx = fmaxf(mx, am[i][k]);
        float sum = 0.f;
        for (int j = 0; j < NN; ++j) { float e = expf(sim[i][j] - mx); sim[i][j] = e; sum += e; }
        for (int k = 0; k < KK; ++k) { float e = expf(am[i][k] - mx); am[i][k] = e; sum += e; }
        float inv = 1.f / sum;
        for (int j = 0; j < NN; ++j) sim[i][j] *= inv;
        for (int k = 0; k < KK; ++k) am[i][k] *= inv;
    }
    __syncthreads();
    for (int t = tid; t < NN * DH; t += 128) {
        int i = t >> 6, d = t & 63;
        float s = 0.f;
        for (int j = 0; j <= i; ++j) s += sim[i][j] * vv[j][d];
        for (int k = 0; k < KK; ++k)
            s += am[i][k] * memv[((size_t)b * MM + ami[i][k]) * DH + d];
        ao[((size_t)b * NN + i) * DD + h * DH + d] = s;
    }
}

// ---------------- memory scores + per-chunk top-32 ----------------
// grid: (chunk=32, tile=15, b=32), block 256 (8 waves).
// q tile (16x64) async-staged to LDS once per block; scores for 16 q-rows x 512 mem
// columns via fp32 WMMA, staged in LDS, then per-row top-32 via wave32 shuffles.
__global__ void k_mscore(const float* __restrict__ qkv, const float* __restrict__ memk,
                         const unsigned char* __restrict__ valid,
                         float* __restrict__ ps, int* __restrict__ pi) {
    int chunk = blockIdx.x;   // 0..31 (512 cols each)
    int tile  = blockIdx.y;   // 0..14 (16 q-rows each, rows = h*20+i)
    int b     = blockIdx.z;   // 0..31
    int tid = threadIdx.x;
    int w = tid >> 5;
    int lane = tid & 31;
    int lr = lane & 15;
    int hi = lane >> 4;

    __shared__ float sc[16][512];               // 32 KB
    __shared__ __align__(16) float q_sh[16][64]; // 4 KB

    // async-stage q tile: 256 x 16B segments, 1 per thread
    {
        int r = tid >> 4;                 // row in tile
        int e4 = (tid & 15) * 4;
        int grow = tile * 16 + r;         // global q-row = h*20 + i
        int h = grow / NN;
        int i = grow % NN;
        async_b128_to_lds(qkv + ((size_t)b * NN + i) * QKVW + h * DH + e4, &q_sh[r][e4]);
    }
    wait_async0();
    __syncthreads();

    for (int ct = w; ct < 32; ct += 8) {
        int j0 = ct * 16;
        int j = chunk * 512 + j0 + lr;
        const float* bbase = memk + ((size_t)b * MM + j) * DH + 2 * hi;
        v8f acc = {};
#pragma unroll 4
        for (int k0 = 0; k0 < DH; k0 += 4) {
            v2f a = *(const v2f*)(&q_sh[lr][k0 + 2 * hi]);
            v2f bf = *(const v2f*)(bbase + k0);
            acc = __builtin_amdgcn_wmma_f32_16x16x4_f32(false, a, false, bf, (short)0, acc,
                                                        false, false);
        }
#pragma unroll
        for (int v = 0; v < 8; ++v) {
            int r = v + 8 * hi;
            int coll = j0 + lr;
            int jg = chunk * 512 + coll;
            sc[r][coll] = valid[(size_t)b * MM + jg] ? acc[v] * 0.125f : -1e9f;
        }
    }
    __syncthreads();

    // top-32 per row, 2 rows per wave
    for (int r = w; r < 16; r += 8) {
        float vals[16];
#pragma unroll
        for (int t = 0; t < 16; ++t) vals[t] = sc[r][lane + 32 * t];
        size_t base = ((((size_t)b * 15 + tile) * 32 + chunk) * 16 + r) * 32;
        for (int it = 0; it < 32; ++it) {
            float ms = -3e38f;
            int mi = 0;
#pragma unroll
            for (int t = 0; t < 16; ++t) {
                if (vals[t] > ms) { ms = vals[t]; mi = lane + 32 * t; }
            }
            for (int off = 16; off > 0; off >>= 1) {
                float os = __shfl_xor(ms, off);
                int oi = __shfl_xor(mi, off);
                if (os > ms || (os == ms && oi < mi)) { ms = os; mi = oi; }
            }
            if ((mi & 31) == lane) vals[mi >> 5] = -3e38f;
            if (lane == 0) { ps[base + it] = ms; pi[base + it] = chunk * 512 + mi; }
        }
    }
}

// ---------------- merge per-chunk top-32 (32 chunks x 32 -> top-32) ----------------
__global__ void k_topk_merge(const float* __restrict__ ps, const int* __restrict__ pi,
                             float* __restrict__ ts, int* __restrict__ ti) {
    int rowid = blockIdx.x;          // 0 .. BB*240-1
    int b = rowid / 240;
    int g = rowid % 240;
    int tile = g >> 4;
    int r = g & 15;
    int tid = threadIdx.x;           // 256
    __shared__ float cs[1024];
    __shared__ int ci[1024];
    __shared__ float rv[256];
    __shared__ int ri[256];
    for (int t = tid; t < 1024; t += 256) {
        int chunk = t >> 5, slot = t & 31;
        size_t idx = ((((size_t)b * 15 + tile) * 32 + chunk) * 16 + r) * 32 + slot;
        cs[t] = ps[idx];
        ci[t] = pi[idx];
    }
    __syncthreads();
    size_t obase = (size_t)rowid * 32;
    for (int it = 0; it < 32; ++it) {
        float ms = -3e38f;
        int mslot = 0;
        for (int t = tid; t < 1024; t += 256) {
            if (cs[t] > ms) { ms = cs[t]; mslot = t; }
        }
        rv[tid] = ms; ri[tid] = mslot;
        __syncthreads();
        for (int off = 128; off > 0; off >>= 1) {
            if (tid < off) {
                if (rv[tid + off] > rv[tid] ||
                    (rv[tid + off] == rv[tid] && ri[tid + off] < ri[tid])) {
                    rv[tid] = rv[tid + off];
                    ri[tid] = ri[tid + off];
                }
            }
            __syncthreads();
        }
        if (tid == 0) {
            ts[obase + it] = rv[0];
            ti[obase + it] = ci[ri[0]];
            cs[ri[0]] = -3e38f;
        }
        __syncthreads();
    }
}

// ---------------- host launcher ----------------
extern "C" void kernel_launch(void* const* d_in, const int* in_sizes, int n_in,
                              void* d_out, int out_size, void* d_ws, size_t ws_size,
                              hipStream_t stream) {
    const float* x      = (const float*)d_in[0];
    const float* W_lin  = (const float*)d_in[1];
    const float* b_lin  = (const float*)d_in[2];
    const float* prefix = (const float*)d_in[3];
    const float* ln1g   = (const float*)d_in[4];
    const float* ln1b   = (const float*)d_in[5];
    const float* ln2g   = (const float*)d_in[6];
    const float* ln2b   = (const float*)d_in[7];
    const float* Wqkv   = (const float*)d_in[8];
    const float* Wo     = (const float*)d_in[9];
    const float* Wff1   = (const float*)d_in[10];
    const float* Wff2   = (const float*)d_in[11];
    const float* lnfg   = (const float*)d_in[12];
    const float* lnfb   = (const float*)d_in[13];
    const float* memk   = (const float*)d_in[14];
    const float* memv   = (const float*)d_in[15];
    const unsigned char* mvalid = (const unsigned char*)d_in[16];
    float* out = (float*)d_out;

    // workspace layout (floats)
    float* ws = (float*)d_ws;
    size_t off = 0;
    float* h      = ws + off; off += (size_t)BB * CLIPL * DD;         // 245760
    float* seq    = ws + off; off += (size_t)BB * NN * DD;            // 491520
    float* y      = ws + off; off += (size_t)BB * NN * DD;            // 491520
    float* qkv    = ws + off; off += (size_t)BB * NN * QKVW;          // 1474560
    float* ao     = ws + off; off += (size_t)BB * NN * DD;            // 491520
    float* ffh    = ws + off; off += (size_t)BB * NN * DFF;           // 1966080
    float* ts     = ws + off; off += (size_t)BB * HH * NN * KK;       // 245760
    int*   ti     = (int*)(ws + off); off += (size_t)BB * HH * NN * KK;
    float* psbuf  = ws + off; off += (size_t)BB * 15 * 32 * 16 * 32;  // 7864320
    int*   pibuf  = (int*)(ws + off); off += (size_t)BB * 15 * 32 * 16 * 32;

    auto gemm = [&](const float* A, const float* B, const float* bias, const float* resid,
                    float* C, int Mr, int Kd, int Nd, int mode) {
        dim3 grid(Nd / 64, Mr / 16);
        k_gemm_f32_wmma<<<grid, dim3(128), 0, stream>>>(A, B, bias, resid, C, Mr, Kd, Nd, mode);
    };

    // 1. h = x @ W_lin + b_lin   (32 x 512 -> 7680)
    gemm(x, W_lin, b_lin, nullptr, h, BB, 512, CLIPL * DD, MODE_BIAS);

    // 2. seq = concat(h.reshape(B,10,D), prefix broadcast)
    {
        int total = BB * NN * DD;
        k_build_seq<<<(total + 255) / 256, 256, 0, stream>>>(h, prefix, seq);
    }

    // 3. transformer layers
    for (int l = 0; l < 8; ++l) {
        // y = LN1(seq)
        k_ln<<<BB * NN, 256, 0, stream>>>(seq, ln1g + (size_t)l * DD, ln1b + (size_t)l * DD, y);
        // qkv = y @ Wqkv[l]
        gemm(y, Wqkv + (size_t)l * DD * QKVW, nullptr, nullptr, qkv, BB * NN, DD, QKVW, MODE_STORE);

        if (l == 4) {
            // memory scores + chunk top-32
            k_mscore<<<dim3(32, 15, BB), 256, 0, stream>>>(qkv, memk, mvalid, psbuf, pibuf);
            // merge to global top-32
            k_topk_merge<<<BB * 240, 256, 0, stream>>>(psbuf, pibuf, ts, ti);
            // attention with memory
            k_attn_mem<<<BB * HH, 128, 0, stream>>>(qkv, ts, ti, memv, ao);
        } else {
            k_attn<<<BB * HH, 128, 0, stream>>>(qkv, ao);
        }

        // seq = seq + ao @ Wo[l]
        gemm(ao, Wo + (size_t)l * DD * DD, nullptr, seq, seq, BB * NN, DD, DD, MODE_RESID);
        // y = LN2(seq)
        k_ln<<<BB * NN, 256, 0, stream>>>(seq, ln2g + (size_t)l * DD, ln2b + (size_t)l * DD, y);
        // ffh = gelu(y @ Wff1[l])
        gemm(y, Wff1 + (size_t)l * DD * DFF, nullptr, nullptr, ffh, BB * NN, DD, DFF, MODE_GELU);
        // seq = seq + ffh @ Wff2[l]
        gemm(ffh, Wff2 + (size_t)l * DFF * DD, nullptr, seq, seq, BB * NN, DFF, DD, MODE_RESID);
    }

    // 4. out = LN_f(seq)[:, 10:, :]
    k_final_ln<<<BB * CLIPL, 256, 0, stream>>>(seq, lnfg, lnfb, out);
}